// PointCloudLoss_18537078850039
// MI455X (gfx1250) — compile-verified
//
#include <hip/hip_runtime.h>

#define BATCH 8
#define NPTS 2048
#define MPTS 2048
#define EPSV 1e-3f
#define INV_EPS 1000.0f
#define SINK_ITERS 50
#define NEGBIG -1e9f
#define BIGF 1e10f
#define LOG2E 1.4426950408889634f
#define LN2F 0.6931471805599453f
#define KSCALE (INV_EPS * LOG2E)     /* 1442.695... : folds 1/eps and base-2 */

typedef float v2f __attribute__((ext_vector_type(2)));
typedef float v8f __attribute__((ext_vector_type(8)));

// ---------------------------------------------------------------------------
// Prep: clip to [-1,1], pack (x,y,z,|p|^2) float4, base-2 log-marginals,
// zero duals, Q = log2-marginal + K*dual (dual=0), init min/cost buffers.
// ---------------------------------------------------------------------------
__global__ void pcl_prep(const float* __restrict__ pred, const float* __restrict__ gt,
                         const int* __restrict__ nlen, const int* __restrict__ mlen,
                         float4* __restrict__ X4, float4* __restrict__ Y4,
                         float* __restrict__ LOGA2, float* __restrict__ LOGB2,
                         float* __restrict__ QA, float* __restrict__ QB,
                         float* __restrict__ F, float* __restrict__ G,
                         float* __restrict__ DX, unsigned int* __restrict__ DY,
                         float* __restrict__ EMD)
{
    const int b = blockIdx.y;
    const int i = blockIdx.x * blockDim.x + threadIdx.x;
    if (i >= NPTS) return;
    {
        const float* p = pred + ((size_t)b * NPTS + i) * 3;
        float x0 = fminf(fmaxf(p[0], -1.f), 1.f);
        float x1 = fminf(fmaxf(p[1], -1.f), 1.f);
        float x2 = fminf(fmaxf(p[2], -1.f), 1.f);
        X4[b * NPTS + i] = make_float4(x0, x1, x2, x0 * x0 + x1 * x1 + x2 * x2);
        int L = nlen[b];
        float la2 = (i < L) ? -__log2f((float)L) : NEGBIG * LOG2E;
        LOGA2[b * NPTS + i] = la2;
        QA[b * NPTS + i] = la2;            // + K*F, F==0
        F[b * NPTS + i] = 0.f;
        DX[b * NPTS + i] = BIGF;
    }
    {
        const float* p = gt + ((size_t)b * MPTS + i) * 3;
        float y0 = fminf(fmaxf(p[0], -1.f), 1.f);
        float y1 = fminf(fmaxf(p[1], -1.f), 1.f);
        float y2 = fminf(fmaxf(p[2], -1.f), 1.f);
        Y4[b * MPTS + i] = make_float4(y0, y1, y2, y0 * y0 + y1 * y1 + y2 * y2);
        int L = mlen[b];
        float lb2 = (i < L) ? -__log2f((float)L) : NEGBIG * LOG2E;
        LOGB2[b * MPTS + i] = lb2;
        QB[b * MPTS + i] = lb2;            // + K*G, G==0
        G[b * MPTS + i] = 0.f;
        DY[b * MPTS + i] = __float_as_uint(BIGF);
    }
    if (b == 0 && blockIdx.x == 0 && i < BATCH) EMD[i] = 0.f;
}

// ---------------------------------------------------------------------------
// Chamfer via V_WMMA_F32_16X16X4_F32. One wave per 16-row strip.
// A (16x4): K0..2 = x, K3 = |x|^2 ; B (4x16): K0..2 = -2*y, K3 = 1
// C/D layout: VGPR i = row n0 + i + 8*(lane/16), col = m0 + lane%16.
// ---------------------------------------------------------------------------
__global__ void pcl_chamfer_wmma(const float4* __restrict__ X4, const float4* __restrict__ Y4,
                                 const int* __restrict__ nlen, const int* __restrict__ mlen,
                                 float* __restrict__ DX, unsigned int* __restrict__ DY)
{
    const int b = blockIdx.y;
    const int n0 = blockIdx.x * 16;
    const int lane = threadIdx.x;   // 0..31 (wave32)
    const int half = lane >> 4;
    const int l = lane & 15;
    const int Ln = nlen[b];
    const int Lm = mlen[b];

    const float4 xp = X4[b * NPTS + n0 + l];
    v2f a;
    a.x = half ? xp.z : xp.x;   // K2 : K0
    a.y = half ? xp.w : xp.y;   // K3 : K1

    float rm[8];
#pragma unroll
    for (int i = 0; i < 8; ++i) rm[i] = BIGF;

    for (int m0 = 0; m0 < MPTS; m0 += 16) {
        const float4 yp = Y4[b * MPTS + m0 + l];
        v2f bm;
        bm.x = half ? (-2.f * yp.z) : (-2.f * yp.x);
        bm.y = half ? 1.0f          : (-2.f * yp.y);
        v8f cz = {0.f, 0.f, 0.f, 0.f, 0.f, 0.f, 0.f, 0.f};
        v8f d = __builtin_amdgcn_wmma_f32_16x16x4_f32(
            false, a, false, bm, (short)0, cz, false, false);

        const float ynv = yp.w;
        const bool colValid = (m0 + l) < Lm;
        float colmin = BIGF;
#pragma unroll
        for (int i = 0; i < 8; ++i) {
            float dv = fmaxf(d[i] + ynv, 0.f);
            rm[i] = fminf(rm[i], colValid ? dv : BIGF);
            int row = n0 + i + 8 * half;
            colmin = fminf(colmin, (row < Ln) ? dv : BIGF);
        }
        colmin = fminf(colmin, __shfl_xor(colmin, 16, 32));
        if (half == 0)
            atomicMin(&DY[b * MPTS + m0 + l], __float_as_uint(colmin));
    }

#pragma unroll
    for (int i = 0; i < 8; ++i) {
#pragma unroll
        for (int off = 1; off < 16; off <<= 1)
            rm[i] = fminf(rm[i], __shfl_xor(rm[i], off, 32));
    }
    if (l == 0) {
#pragma unroll
        for (int i = 0; i < 8; ++i)
            DX[b * NPTS + n0 + i + 8 * half] = rm[i];
    }
}

// ---------------------------------------------------------------------------
// One Sinkhorn half-iteration, WMMA cost tiles + two-pass (max, then sum)
// base-2 LSE.  t2(n,m) = Qcol[m] - K*|y_m|^2 - acc(n,m),
// acc = WMMA giving K*(|x_n|^2 - 2 x_n.y_m)  (K = log2e/eps folded into B).
// rowDual[n] = -eps*ln2*( mx2 + log2(sum) );  Qrow[n] = log2row[n] + K*f.
// One wave per 16-row strip; 8 per-lane (row i / i+8) max / sum registers,
// shfl_xor butterfly within each 16-lane half for the final row reduction.
// ---------------------------------------------------------------------------
__global__ void pcl_sinkhorn_wmma(const float4* __restrict__ RowPts,
                                  const float4* __restrict__ ColPts,
                                  const float* __restrict__ Qcol,
                                  const float* __restrict__ LOG2row,
                                  float* __restrict__ rowDual,
                                  float* __restrict__ Qrow,
                                  int R, int Cn)
{
    const int b = blockIdx.y;
    const int wave = threadIdx.x >> 5;
    const int lane = threadIdx.x & 31;
    const int half = lane >> 4;
    const int l = lane & 15;
    const int n0 = (blockIdx.x * (blockDim.x >> 5) + wave) * 16;

    const float4* yB = ColPts + (size_t)b * Cn;
    const float* qB = Qcol + (size_t)b * Cn;

    const float4 xp = RowPts[b * R + n0 + l];
    v2f a;
    a.x = half ? xp.z : xp.x;   // K2 : K0
    a.y = half ? xp.w : xp.y;   // K3 : K1

    float mx8[8];
#pragma unroll
    for (int i = 0; i < 8; ++i) mx8[i] = -3.0e38f;

    // ---- pass 1: row maxima of t2 ----
    for (int m0 = 0; m0 < Cn; m0 += 16) {
        const float4 yp = yB[m0 + l];
        if (m0 + 64 < Cn) __builtin_prefetch(&yB[m0 + 64], 0, 1);
        const float qv = qB[m0 + l] - KSCALE * yp.w;
        v2f bm;
        bm.x = half ? (-2.f * KSCALE * yp.z) : (-2.f * KSCALE * yp.x);
        bm.y = half ? KSCALE                 : (-2.f * KSCALE * yp.y);
        v8f cz = {0.f, 0.f, 0.f, 0.f, 0.f, 0.f, 0.f, 0.f};
        v8f acc = __builtin_amdgcn_wmma_f32_16x16x4_f32(
            false, a, false, bm, (short)0, cz, false, false);
#pragma unroll
        for (int i = 0; i < 8; ++i)
            mx8[i] = fmaxf(mx8[i], qv - acc[i]);
    }
#pragma unroll
    for (int i = 0; i < 8; ++i) {
#pragma unroll
        for (int off = 1; off < 16; off <<= 1)
            mx8[i] = fmaxf(mx8[i], __shfl_xor(mx8[i], off, 32));
    }

    // ---- pass 2: row sums of exp2(t2 - mx) ----
    float s8[8];
#pragma unroll
    for (int i = 0; i < 8; ++i) s8[i] = 0.f;

    for (int m0 = 0; m0 < Cn; m0 += 16) {
        const float4 yp = yB[m0 + l];
        const float qv = qB[m0 + l] - KSCALE * yp.w;
        v2f bm;
        bm.x = half ? (-2.f * KSCALE * yp.z) : (-2.f * KSCALE * yp.x);
        bm.y = half ? KSCALE                 : (-2.f * KSCALE * yp.y);
        v8f cz = {0.f, 0.f, 0.f, 0.f, 0.f, 0.f, 0.f, 0.f};
        v8f acc = __builtin_amdgcn_wmma_f32_16x16x4_f32(
            false, a, false, bm, (short)0, cz, false, false);
#pragma unroll
        for (int i = 0; i < 8; ++i)
            s8[i] += __builtin_exp2f(qv - acc[i] - mx8[i]);
    }
#pragma unroll
    for (int i = 0; i < 8; ++i) {
#pragma unroll
        for (int off = 1; off < 16; off <<= 1)
            s8[i] += __shfl_xor(s8[i], off, 32);
    }

    if (l == 0) {   // lanes 0 (rows n0..n0+7) and 16 (rows n0+8..n0+15)
#pragma unroll
        for (int i = 0; i < 8; ++i) {
            const int row = n0 + i + 8 * half;
            const float lse2 = mx8[i] + __log2f(s8[i]);
            const float fnew = -EPSV * LN2F * lse2;
            rowDual[b * R + row] = fnew;
            Qrow[b * R + row] = LOG2row[b * R + row] + KSCALE * fnew;
        }
    }
}

// ---------------------------------------------------------------------------
// Transport cost: sum exp2( la2 + lb2 + (f+g-C)*K ) * C   per batch.
// ---------------------------------------------------------------------------
__global__ void pcl_emd_cost(const float4* __restrict__ X4, const float4* __restrict__ Y4,
                             const float* __restrict__ F, const float* __restrict__ G,
                             const float* __restrict__ LOGA2, const float* __restrict__ LOGB2,
                             float* __restrict__ EMD)
{
    const int b = blockIdx.y;
    const int wave = threadIdx.x >> 5;
    const int lane = threadIdx.x & 31;
    const int row = blockIdx.x * (blockDim.x >> 5) + wave;
    if (row >= NPTS) return;

    const float4 xp = X4[b * NPTS + row];
    const float fr = F[b * NPTS + row];
    const float la = LOGA2[b * NPTS + row];
    const float4* yB = Y4 + (size_t)b * MPTS;
    const float* gB = G + (size_t)b * MPTS;
    const float* lB = LOGB2 + (size_t)b * MPTS;

    float acc = 0.f;
    for (int m = lane; m < MPTS; m += 32) {
        float4 yp = yB[m];
        float C = fmaxf(xp.w + yp.w -
                        2.f * (xp.x * yp.x + xp.y * yp.y + xp.z * yp.z), 0.f);
        float lp2 = la + lB[m] + (fr + gB[m] - C) * KSCALE;
        acc += __builtin_exp2f(lp2) * C;
    }
#pragma unroll
    for (int off = 16; off; off >>= 1) acc += __shfl_xor(acc, off, 32);
    if (lane == 0) atomicAdd(&EMD[b], acc);
}

// ---------------------------------------------------------------------------
// Final scalar: cd = mean_b( sum dx/n + sum dy/m ) ; emd = mean_b sqrt(cost).
// ---------------------------------------------------------------------------
__global__ void pcl_finalize(const float* __restrict__ DX, const unsigned int* __restrict__ DY,
                             const float* __restrict__ EMD,
                             const int* __restrict__ nlen, const int* __restrict__ mlen,
                             float* __restrict__ out)
{
    __shared__ float red[256];
    float acc = 0.f;
    for (int idx = threadIdx.x; idx < BATCH * NPTS; idx += blockDim.x) {
        int b = idx / NPTS, n = idx % NPTS;
        int L = nlen[b];
        if (n < L) acc += DX[idx] / (float)L;
    }
    for (int idx = threadIdx.x; idx < BATCH * MPTS; idx += blockDim.x) {
        int b = idx / MPTS, m = idx % MPTS;
        int L = mlen[b];
        if (m < L) acc += __uint_as_float(DY[idx]) / (float)L;
    }
    red[threadIdx.x] = acc;
    __syncthreads();
    for (int off = 128; off; off >>= 1) {
        if (threadIdx.x < off) red[threadIdx.x] += red[threadIdx.x + off];
        __syncthreads();
    }
    if (threadIdx.x == 0) {
        float cd = red[0] / (float)BATCH;
        float emd = 0.f;
        for (int b = 0; b < BATCH; ++b) emd += sqrtf(fmaxf(EMD[b], 0.f));
        emd /= (float)BATCH;
        out[0] = cd + emd;
    }
}

// ---------------------------------------------------------------------------
extern "C" void kernel_launch(void* const* d_in, const int* in_sizes, int n_in,
                              void* d_out, int out_size, void* d_ws, size_t ws_size,
                              hipStream_t stream)
{
    (void)in_sizes; (void)n_in; (void)out_size; (void)ws_size;
    const float* pred = (const float*)d_in[0];
    const float* gt   = (const float*)d_in[1];
    const int* nlen   = (const int*)d_in[2];
    const int* mlen   = (const int*)d_in[3];
    float* out = (float*)d_out;

    char* ws = (char*)d_ws;
    auto carve = [&](size_t bytes) -> char* {
        char* p = ws;
        ws += (bytes + 255) & ~(size_t)255;
        return p;
    };
    float4* X4 = (float4*)carve((size_t)BATCH * NPTS * sizeof(float4));
    float4* Y4 = (float4*)carve((size_t)BATCH * MPTS * sizeof(float4));
    float* LOGA2 = (float*)carve((size_t)BATCH * NPTS * sizeof(float));
    float* LOGB2 = (float*)carve((size_t)BATCH * MPTS * sizeof(float));
    float* QA = (float*)carve((size_t)BATCH * NPTS * sizeof(float));
    float* QB = (float*)carve((size_t)BATCH * MPTS * sizeof(float));
    float* F = (float*)carve((size_t)BATCH * NPTS * sizeof(float));
    float* G = (float*)carve((size_t)BATCH * MPTS * sizeof(float));
    float* DX = (float*)carve((size_t)BATCH * NPTS * sizeof(float));
    unsigned int* DY = (unsigned int*)carve((size_t)BATCH * MPTS * sizeof(unsigned int));
    float* EMD = (float*)carve(BATCH * sizeof(float));

    pcl_prep<<<dim3(NPTS / 256, BATCH), 256, 0, stream>>>(
        pred, gt, nlen, mlen, X4, Y4, LOGA2, LOGB2, QA, QB, F, G, DX, DY, EMD);

    pcl_chamfer_wmma<<<dim3(NPTS / 16, BATCH), 32, 0, stream>>>(
        X4, Y4, nlen, mlen, DX, DY);

    for (int it = 0; it < SINK_ITERS; ++it) {
        // f-update (uses Q built from g), writes F and QA for the g-update
        pcl_sinkhorn_wmma<<<dim3(NPTS / 64, BATCH), 128, 0, stream>>>(
            X4, Y4, QB, LOGA2, F, QA, NPTS, MPTS);
        // g-update (uses Q built from new f), writes G and QB
        pcl_sinkhorn_wmma<<<dim3(MPTS / 64, BATCH), 128, 0, stream>>>(
            Y4, X4, QA, LOGB2, G, QB, MPTS, NPTS);
    }

    pcl_emd_cost<<<dim3(NPTS / 8, BATCH), 256, 0, stream>>>(
        X4, Y4, F, G, LOGA2, LOGB2, EMD);

    pcl_finalize<<<1, 256, 0, stream>>>(DX, DY, EMD, nlen, mlen, out);
}